// SpectrumExtractionByCheapTrick_69956427317947
// MI455X (gfx1250) — compile-verified
//
#include <hip/hip_runtime.h>
#include <hip/hip_bf16.h>
#include <stdint.h>

typedef __attribute__((ext_vector_type(2))) float v2f;
typedef __attribute__((ext_vector_type(8))) float v8f;

#define L_FFT   2048
#define HALF_L  1024
#define K1      1025
#define FS_F    24000.0f
#define FRAME_P 120
#define T_LEN   120000
#define NFR     1001
#define NFT     8008           /* total frames = 8*1001 */
#define NFT_PAD 8192           /* 512 m-tiles of 16 rows */
#define MBLK    16             /* m-blocks; each covers 32 m-tiles (512 rows) */
#define NPAD    1040           /* 65*16 */
#define NTILES  65
#define KGF     512            /* 2048/4 k-groups for FFT GEMM */
#define KGC     257            /* 1028/4 k-groups for cosine GEMMs */
#define NC      32             /* k-groups staged per LDS chunk */
#define CH_F    16             /* KGF/NC */
#define CH_C    9              /* ceil(KGC/NC) */
#define MREP    4              /* m-tiles per wave */
#define KPAD_SM 36
#define MSM     1097           /* K1 + 2*KPAD_SM */
#define F_MIN_V 35.207824f     /* 3*24000/2045 */
#define EPS_F32V 1.1920929e-7f
#define PI_F    3.14159265358979f

// ---------------------------------------------------------------------------
// f32 WMMA 16x16x4 (CDNA5). A:16x4 (v2f/lane), B:4x16 (v2f/lane), C/D v8f.
// A: lanes 0-15 -> M=lane, K=k0+{0,1}; lanes 16-31 -> M=lane-16, K=k0+{2,3}
// B: lanes 0-15 -> N=lane,  K=k0+{0,1}; lanes 16-31 -> N=lane-16, K=k0+{2,3}
// D: VGPR i: lanes 0-15 -> M=i,N=lane; lanes 16-31 -> M=i+8,N=lane-16
// ---------------------------------------------------------------------------
__device__ __forceinline__ v8f wmma4(v2f a, v2f b, v8f c) {
  return __builtin_amdgcn_wmma_f32_16x16x4_f32(
      false, a, false, b, (short)0, c, false, false);
}

__device__ __forceinline__ float fixf0(float f) {
  return (f <= F_MIN_V) ? 500.0f : f;
}

// Async global->LDS copy (ASYNCcnt-tracked), per CDNA5 ISA.
// dst operand is the per-lane LDS byte address; generic shared pointers keep
// the LDS offset in addr[31:0] (ISA 10.2 aperture rules), so truncation works.
__device__ __forceinline__ void async_b64(unsigned lds_addr, const float* g) {
  asm volatile("global_load_async_to_lds_b64 %0, %1, off"
               :: "v"(lds_addr), "v"(g) : "memory");
}
__device__ __forceinline__ void wait_async0() {
  asm volatile("s_wait_asynccnt 0x0" ::: "memory");
}
// n64 = number of 8-byte packets; 256 threads cooperate.
__device__ __forceinline__ void stage_async(float* dst, const float* src,
                                            int n64, int tid) {
  for (int i = tid; i < n64; i += 256)
    async_b64((unsigned)(uintptr_t)(dst + i * 2), src + (size_t)i * 2);
}

// ---------------------------------------------------------------------------
// Stage 0: WMMA-packed basis tables.
//   Fc/Fs: FFT cos / -sin bases (K=time t 2048, N=bin 1025 pad 1040)
//   Mt   : w_k * cos(2*pi*k*n/L); used for irfft (1/L in epilogue) and hfft.
// Packed index: ((ntile*KG + kg)*32 + lane)*2 + j,
//   n = ntile*16 + (lane&15);  k = kg*4 + (lane>>4)*2 + j
// ---------------------------------------------------------------------------
__global__ void __launch_bounds__(256) ktab(float* __restrict__ Fc,
                                            float* __restrict__ Fs,
                                            float* __restrict__ Mt) {
  long idx = (long)blockIdx.x * 256 + threadIdx.x;
  const long totF = (long)NTILES * KGF * 64;
  if (idx >= totF) return;
  int j    = (int)(idx & 1);
  int lane = (int)((idx >> 1) & 31);
  long e   = idx >> 6;
  int kg   = (int)(e % KGF);
  int nt   = (int)(e / KGF);
  int n    = nt * 16 + (lane & 15);
  int k    = kg * 4 + (lane >> 4) * 2 + j;

  float c = 0.0f, s = 0.0f;
  if (n < K1) {
    int ph = (n * k) & (L_FFT - 1);            // exact phase mod L
    float th = (float)ph * (PI_F / (float)HALF_L);
    c = cosf(th);
    s = -sinf(th);
  }
  Fc[idx] = c;
  Fs[idx] = s;

  if (kg < KGC) {
    long midx = (((long)nt * KGC + kg) * 32 + lane) * 2 + j;
    float mv = 0.0f;
    if (n < K1 && k < K1) {
      float w = (k == 0 || k == HALF_L) ? 1.0f : 2.0f;
      int ph = (n * k) & (L_FFT - 1);
      mv = w * cosf((float)ph * (PI_F / (float)HALF_L));
    }
    Mt[midx] = mv;
  }
}

// ---------------------------------------------------------------------------
// Stage 1: framing + window + normalization + weighted mean removal.
// wav[fr][t] = win_n[t] * (frame[t] - sum(f*w)/sum(w))
// ---------------------------------------------------------------------------
__global__ void __launch_bounds__(256) kwin(const float* __restrict__ x,
                                            const float* __restrict__ f0g,
                                            float* __restrict__ wav) {
  __shared__ float sf[L_FFT];
  __shared__ float sw[L_FFT];
  __shared__ float r0[256], r1[256], r2[256];
  int fr = blockIdx.x;
  int b  = fr / NFR;
  int n  = fr - b * NFR;
  int tid = threadIdx.x;
  float f0 = fixf0(f0g[fr]);
  float hwl = rintf(36000.0f / f0);            // round(1.5*fs/f0), half-even
  const float* xb = x + (size_t)b * T_LEN;
  int center = n * FRAME_P;

  float a0 = 0.0f, a1 = 0.0f, a2 = 0.0f;
  for (int t = tid; t < L_FFT; t += 256) {
    int rel = t - HALF_L;
    int ix = center + rel;
    ix = ix < 0 ? 0 : (ix > T_LEN - 1 ? T_LEN - 1 : ix);
    float fv = xb[ix];
    float m = (fabsf((float)rel) <= hwl) ? 1.0f : 0.0f;
    float w = (0.5f * cosf((PI_F / 36000.0f) * (float)rel * f0) + 0.5f) * m;
    sf[t] = fv;
    sw[t] = w;
    a0 += w * w; a1 += w; a2 += fv * w;
  }
  r0[tid] = a0; r1[tid] = a1; r2[tid] = a2;
  __syncthreads();
  for (int off = 128; off > 0; off >>= 1) {
    if (tid < off) {
      r0[tid] += r0[tid + off];
      r1[tid] += r1[tid + off];
      r2[tid] += r2[tid + off];
    }
    __syncthreads();
  }
  float scale = rsqrtf(r0[0]);
  float ratio = r2[0] / r1[0];
  float* wrow = wav + (size_t)fr * L_FFT;
  for (int t = tid; t < L_FFT; t += 256)
    wrow[t] = sw[t] * scale * (sf[t] - ratio);
}

// ---------------------------------------------------------------------------
// Stage 2: power spectrum GEMM. 8 waves/block share one N-tile; the cos/sin
// B-stream is async-staged into double-buffered LDS; each wave register-
// blocks 4 M-tiles (8 f32 WMMA accumulators).
// ---------------------------------------------------------------------------
__global__ void __launch_bounds__(256) kfft(const float* __restrict__ wav,
                                            const float* __restrict__ Fc,
                                            const float* __restrict__ Fs,
                                            float* __restrict__ P) {
  __shared__ float sB[2][2][NC * 64];          // [buf][cos/sin] : 32 KB
  int tid  = threadIdx.x;
  int wid  = tid >> 5;
  int lane = tid & 31;
  int nt = blockIdx.x / MBLK;
  int mb = blockIdx.x % MBLK;
  int hw = lane >> 4, l16 = lane & 15;

  const float* ap[MREP];
#pragma unroll
  for (int r = 0; r < MREP; ++r) {
    int arow = (mb * 32 + wid * MREP + r) * 16 + l16;
    if (arow > NFT - 1) arow = NFT - 1;        // clamp padded rows
    ap[r] = wav + (size_t)arow * L_FFT + hw * 2;
  }
  const float* bcg = Fc + (size_t)nt * KGF * 64;
  const float* bsg = Fs + (size_t)nt * KGF * 64;

  v8f cre[MREP], cim[MREP];
#pragma unroll
  for (int r = 0; r < MREP; ++r) { cre[r] = (v8f){0,0,0,0,0,0,0,0};
                                   cim[r] = (v8f){0,0,0,0,0,0,0,0}; }

  stage_async(&sB[0][0][0], bcg, NC * 32, tid);
  stage_async(&sB[0][1][0], bsg, NC * 32, tid);
  wait_async0();
  __syncthreads();

  for (int c = 0; c < CH_F; ++c) {
    int cur = c & 1;
    if (c + 1 < CH_F) {
      stage_async(&sB[cur ^ 1][0][0], bcg + (size_t)(c + 1) * NC * 64, NC * 32, tid);
      stage_async(&sB[cur ^ 1][1][0], bsg + (size_t)(c + 1) * NC * 64, NC * 32, tid);
    }
    const float* Bc = &sB[cur][0][0];
    const float* Bs = &sB[cur][1][0];
#pragma unroll 4
    for (int kg = 0; kg < NC; ++kg) {
      v2f b0 = *(const v2f*)(Bc + kg * 64 + lane * 2);
      v2f b1 = *(const v2f*)(Bs + kg * 64 + lane * 2);
      int kk = (c * NC + kg) * 4;
#pragma unroll
      for (int r = 0; r < MREP; ++r) {
        v2f a = *(const v2f*)(ap[r] + kk);
        cre[r] = wmma4(a, b0, cre[r]);
        cim[r] = wmma4(a, b1, cim[r]);
      }
    }
    wait_async0();
    __syncthreads();
  }

  int col = nt * 16 + l16;
#pragma unroll
  for (int r = 0; r < MREP; ++r) {
    float* prow = P + (size_t)((mb * 32 + wid * MREP + r) * 16 + hw * 8) * NPAD + col;
#pragma unroll
    for (int i = 0; i < 8; ++i) {
      float re = cre[r][i], im = cim[r][i];
      prow[(size_t)i * NPAD] = re * re + im * im;
    }
  }
}

// ---------------------------------------------------------------------------
// Stage 3: per-frame mirror replacement + reflected cumsum smoothing + log.
// ---------------------------------------------------------------------------
__device__ __forceinline__ float interpC(const float* C, float pos) {
  int lo = (int)floorf(pos);
  lo = lo < 0 ? 0 : (lo > MSM - 2 ? MSM - 2 : lo);
  float f = pos - (float)lo;
  return C[lo] + (C[lo + 1] - C[lo]) * f;
}

__global__ void __launch_bounds__(256) ksm(const float* __restrict__ f0g,
                                           float* __restrict__ PL) {
  __shared__ float sP[K1];
  __shared__ float sC[MSM];
  __shared__ float part[256];
  int fr = blockIdx.x, tid = threadIdx.x;
  float f0 = fixf0(f0g[fr]);
  float rate = f0 * (float)L_FFT / FS_F;
  float* row = PL + (size_t)fr * NPAD;

  for (int k = tid; k < K1; k += 256) sP[k] = row[k];
  __syncthreads();

  // mirror replacement: P[k] += interp(P, rate-k) for k < 1+floor(rate)
  int kmax = 1 + (int)floorf(rate);
  float addv[5];
  int c = 0;
  for (int k = tid; k < K1; k += 256) {
    float add = 0.0f;
    if (k < kmax) {
      float m = rate - (float)k;
      int lo = (int)floorf(m);
      lo = lo < 0 ? 0 : (lo > K1 - 2 ? K1 - 2 : lo);
      float f = m - (float)lo;
      add = sP[lo] * (1.0f - f) + sP[lo + 1] * f;
    }
    addv[c++] = add;
  }
  __syncthreads();
  c = 0;
  for (int k = tid; k < K1; k += 256) sP[k] += addv[c++];
  __syncthreads();

  // inclusive cumsum of reflected, scaled spectrum (1097 entries)
  const float scl = FS_F / (float)L_FFT;
  int base = tid * 5;
  float run = 0.0f;
  for (int q = 0; q < 5; ++q) {
    int jj = base + q;
    if (jj < MSM) {
      int i = jj - KPAD_SM;
      int a = i < 0 ? -i : i;
      if (a > HALF_L) a = L_FFT - a;
      run += sP[a] * scl;
      sC[jj] = run;
    }
  }
  part[tid] = run;
  __syncthreads();
  for (int off = 1; off < 256; off <<= 1) {
    float v = (tid >= off) ? part[tid - off] : 0.0f;
    __syncthreads();
    part[tid] += v;
    __syncthreads();
  }
  float excl = (tid > 0) ? part[tid - 1] : 0.0f;
  for (int q = 0; q < 5; ++q) {
    int jj = base + q;
    if (jj < MSM) sC[jj] += excl;
  }
  __syncthreads();

  float width = f0 * (2.0f / 3.0f);
  float wbins = width * (float)L_FFT / FS_F;
  for (int k = tid; k < NPAD; k += 256) {
    float v = 0.0f;
    if (k < K1) {
      float plo = (float)k - 0.5f * wbins + ((float)KPAD_SM - 0.5f);
      float phi = plo + wbins;
      v = (interpC(sC, phi) - interpC(sC, plo)) / width
          + 0.7978845608f * EPS_F32V;          // E[|N|]*eps regularizer
      v = logf(v);
    }
    row[k] = v;
  }
}

// ---------------------------------------------------------------------------
// Stages 4/5 share one GEMM skeleton: async-staged Mt B-stream in LDS,
// 4 M-tiles per wave, different epilogues.
// ---------------------------------------------------------------------------
__global__ void __launch_bounds__(256) kcep(const float* __restrict__ PL,
                                            const float* __restrict__ Mt,
                                            const float* __restrict__ f0g,
                                            float* __restrict__ gcep) {
  __shared__ float sB[2][NC * 64];             // 16 KB
  int tid  = threadIdx.x;
  int wid  = tid >> 5;
  int lane = tid & 31;
  int nt = blockIdx.x / MBLK;
  int mb = blockIdx.x % MBLK;
  int hw = lane >> 4, l16 = lane & 15;

  const float* ap[MREP];
#pragma unroll
  for (int r = 0; r < MREP; ++r) {
    int arow = (mb * 32 + wid * MREP + r) * 16 + l16;
    if (arow > NFT - 1) arow = NFT - 1;
    ap[r] = PL + (size_t)arow * NPAD + hw * 2;
  }
  const float* bg = Mt + (size_t)nt * KGC * 64;

  v8f acc[MREP];
#pragma unroll
  for (int r = 0; r < MREP; ++r) acc[r] = (v8f){0,0,0,0,0,0,0,0};

  int cnt0 = NC;                                // first chunk is full
  stage_async(&sB[0][0], bg, cnt0 * 32, tid);
  wait_async0();
  __syncthreads();

  for (int c = 0; c < CH_C; ++c) {
    int cur = c & 1;
    int cnt = (c * NC + NC <= KGC) ? NC : (KGC - c * NC);
    if (c + 1 < CH_C) {
      int cn = ((c + 1) * NC + NC <= KGC) ? NC : (KGC - (c + 1) * NC);
      stage_async(&sB[cur ^ 1][0], bg + (size_t)(c + 1) * NC * 64, cn * 32, tid);
    }
    const float* Bb = &sB[cur][0];
#pragma unroll 4
    for (int kg = 0; kg < cnt; ++kg) {
      v2f b = *(const v2f*)(Bb + kg * 64 + lane * 2);
      int kk = (c * NC + kg) * 4;
#pragma unroll
      for (int r = 0; r < MREP; ++r) {
        v2f a = *(const v2f*)(ap[r] + kk);
        acc[r] = wmma4(a, b, acc[r]);
      }
    }
    wait_async0();
    __syncthreads();
  }

  int col = nt * 16 + l16;
#pragma unroll
  for (int r = 0; r < MREP; ++r) {
#pragma unroll
    for (int i = 0; i < 8; ++i) {
      int grow = (mb * 32 + wid * MREP + r) * 16 + hw * 8 + i;
      int rr = grow > NFT - 1 ? NFT - 1 : grow;
      float f0 = fixf0(f0g[rr]);
      float v = 0.0f;
      if (col < K1) {
        float z  = f0 * (float)col / FS_F;
        float pz = PI_F * z;
        float sl = (col == 0) ? 1.0f : (sinf(pz) / pz);
        float cl = 1.3f - 0.3f * cosf(2.0f * pz);  // 1-2*Q1+2*Q1*cos, Q1=-0.15
        v = acc[r][i] * sl * cl * (1.0f / (float)L_FFT);
      }
      gcep[(size_t)grow * NPAD + col] = v;
    }
  }
}

__global__ void __launch_bounds__(256) khfft(const float* __restrict__ gcep,
                                             const float* __restrict__ Mt,
                                             float* __restrict__ out) {
  __shared__ float sB[2][NC * 64];
  int tid  = threadIdx.x;
  int wid  = tid >> 5;
  int lane = tid & 31;
  int nt = blockIdx.x / MBLK;
  int mb = blockIdx.x % MBLK;
  int hw = lane >> 4, l16 = lane & 15;

  const float* ap[MREP];
#pragma unroll
  for (int r = 0; r < MREP; ++r) {
    int arow = (mb * 32 + wid * MREP + r) * 16 + l16;
    if (arow > NFT - 1) arow = NFT - 1;
    ap[r] = gcep + (size_t)arow * NPAD + hw * 2;
  }
  const float* bg = Mt + (size_t)nt * KGC * 64;

  v8f acc[MREP];
#pragma unroll
  for (int r = 0; r < MREP; ++r) acc[r] = (v8f){0,0,0,0,0,0,0,0};

  stage_async(&sB[0][0], bg, NC * 32, tid);
  wait_async0();
  __syncthreads();

  for (int c = 0; c < CH_C; ++c) {
    int cur = c & 1;
    int cnt = (c * NC + NC <= KGC) ? NC : (KGC - c * NC);
    if (c + 1 < CH_C) {
      int cn = ((c + 1) * NC + NC <= KGC) ? NC : (KGC - (c + 1) * NC);
      stage_async(&sB[cur ^ 1][0], bg + (size_t)(c + 1) * NC * 64, cn * 32, tid);
    }
    const float* Bb = &sB[cur][0];
#pragma unroll 4
    for (int kg = 0; kg < cnt; ++kg) {
      v2f b = *(const v2f*)(Bb + kg * 64 + lane * 2);
      int kk = (c * NC + kg) * 4;
#pragma unroll
      for (int r = 0; r < MREP; ++r) {
        v2f a = *(const v2f*)(ap[r] + kk);
        acc[r] = wmma4(a, b, acc[r]);
      }
    }
    wait_async0();
    __syncthreads();
  }

  int col = nt * 16 + l16;
  if (col < K1) {
#pragma unroll
    for (int r = 0; r < MREP; ++r) {
#pragma unroll
      for (int i = 0; i < 8; ++i) {
        int grow = (mb * 32 + wid * MREP + r) * 16 + hw * 8 + i;
        if (grow < NFT) out[(size_t)grow * K1 + col] = acc[r][i];
      }
    }
  }
}

// ---------------------------------------------------------------------------
extern "C" void kernel_launch(void* const* d_in, const int* in_sizes, int n_in,
                              void* d_out, int out_size, void* d_ws, size_t ws_size,
                              hipStream_t stream) {
  const float* x  = (const float*)d_in[0];     // (8, 120000) f32
  const float* f0 = (const float*)d_in[1];     // (8, 1001)  f32
  float* out = (float*)d_out;                  // (8, 1001, 1025) f32

  float* ws   = (float*)d_ws;
  float* wav  = ws;                                        // NFT_PAD * 2048
  float* PL   = wav  + (size_t)NFT_PAD * L_FFT;            // NFT_PAD * 1040
  float* gcep = PL   + (size_t)NFT_PAD * NPAD;             // NFT_PAD * 1040
  float* Fc   = gcep + (size_t)NFT_PAD * NPAD;             // 65*512*64
  float* Fs   = Fc   + (size_t)NTILES * KGF * 64;          // 65*512*64
  float* Mt   = Fs   + (size_t)NTILES * KGF * 64;          // 65*257*64

  long tabN = (long)NTILES * KGF * 64;
  ktab<<<(int)((tabN + 255) / 256), 256, 0, stream>>>(Fc, Fs, Mt);
  kwin<<<NFT, 256, 0, stream>>>(x, f0, wav);
  int gblocks = NTILES * MBLK;                 // 1040 blocks, exact grid
  kfft<<<gblocks, 256, 0, stream>>>(wav, Fc, Fs, PL);
  ksm<<<NFT, 256, 0, stream>>>(f0, PL);
  kcep<<<gblocks, 256, 0, stream>>>(PL, Mt, f0, gcep);
  khfft<<<gblocks, 256, 0, stream>>>(gcep, Mt, out);
}